// SingleVolumeRenderer_18296560681710
// MI455X (gfx1250) — compile-verified
//
#include <hip/hip_runtime.h>
#include <hip/hip_bf16.h>

#define CAP_SAMPLES 7168   // staged samples per block: 2 * 7168 * 4B = 57,344 B LDS

#define __AS1 __attribute__((address_space(1)))
#define __AS3 __attribute__((address_space(3)))

typedef int   v4i_t __attribute__((vector_size(4 * sizeof(int))));
typedef float v4f_t __attribute__((vector_size(4 * sizeof(float))));

#if defined(__has_builtin)
#  if __has_builtin(__builtin_amdgcn_global_load_async_to_lds_b128)
#    define HAVE_ASYNC_B128 1
#  else
#    define HAVE_ASYNC_B128 0
#  endif
#  if __has_builtin(__builtin_amdgcn_global_load_async_to_lds_b32)
#    define HAVE_ASYNC_B32 1
#  else
#    define HAVE_ASYNC_B32 0
#  endif
#  if __has_builtin(__builtin_amdgcn_s_wait_asynccnt)
#    define HAVE_WAIT_ASYNC 1
#  else
#    define HAVE_WAIT_ASYNC 0
#  endif
#else
#  define HAVE_ASYNC_B128 0
#  define HAVE_ASYNC_B32 0
#  define HAVE_WAIT_ASYNC 0
#endif

__device__ __forceinline__ void svr_wait_async0() {
#if HAVE_WAIT_ASYNC
  __builtin_amdgcn_s_wait_asynccnt(0);
#else
  asm volatile("s_wait_asynccnt 0x0" ::: "memory");
#endif
}

// Zero the output volumes and the seg_start/seg_end scratch in one launch.
__global__ __launch_bounds__(256) void svr_zero(float* __restrict__ out, int nOut,
                                                int* __restrict__ ws, int nWs) {
  const int i = blockIdx.x * blockDim.x + threadIdx.x;
  const int stride = gridDim.x * blockDim.x;
  const v4f_t z = {0.0f, 0.0f, 0.0f, 0.0f};
  const int n4 = nOut >> 2;
  for (int k = i; k < n4; k += stride) ((v4f_t*)out)[k] = z;
  for (int k = (n4 << 2) + i; k < nOut; k += stride) out[k] = 0.0f;
  for (int k = i; k < nWs; k += stride) ws[k] = 0;
}

// Detect per-segment [start, end) from the sorted per-sample segment ids.
// Block stages its 256-int tile (tile base is 1KB-aligned) into LDS via async
// copies, so each element is read from global exactly once.
__global__ __launch_bounds__(256) void svr_seg_bounds(const int* __restrict__ seg, int T,
                                                      int* __restrict__ segStart,
                                                      int* __restrict__ segEnd) {
  __shared__ int sSeg[257];
  const int i0 = blockIdx.x * 256;
  const int i = i0 + threadIdx.x;
  const int tileN = min(256, T - i0);          // > 0 for every launched block
  const int nFull = tileN >> 2;                // full 16B chunks

#if HAVE_ASYNC_B128
  for (int c = (int)threadIdx.x; c < nFull; c += 256) {
    __builtin_amdgcn_global_load_async_to_lds_b128(
        (__AS1 v4i_t*)(seg + i0 + 4 * c), (__AS3 v4i_t*)(sSeg + 4 * c), 0, 0);
  }
  for (int j = (nFull << 2) + (int)threadIdx.x; j < tileN; j += 256) {
#if HAVE_ASYNC_B32
    __builtin_amdgcn_global_load_async_to_lds_b32(
        (__AS1 int*)(seg + i0 + j), (__AS3 int*)(sSeg + j), 0, 0);
#else
    sSeg[j] = seg[i0 + j];
#endif
  }
  if (threadIdx.x == 0) sSeg[256] = (i0 + 256 < T) ? seg[i0 + 256] : -1;
  svr_wait_async0();
#else
  for (int j = (int)threadIdx.x; j < tileN; j += 256) sSeg[j] = seg[i0 + j];
  if (threadIdx.x == 0) sSeg[256] = (i0 + 256 < T) ? seg[i0 + 256] : -1;
#endif
  __syncthreads();

  if (i >= T) return;
  const int s = sSeg[threadIdx.x];
  const int snext = (i == T - 1) ? -1 : sSeg[threadIdx.x + 1];
  if (i == 0) segStart[s] = 0;
  if (snext != s) {
    segEnd[s] = i + 1;
    if (snext >= 0) segStart[snext] = i + 1;
  }
}

// One wave32 lane per segment; block stages its contiguous alpha/t range in LDS
// via CDNA5 async global->LDS copies when it fits.
__global__ __launch_bounds__(256) void svr_render(
    const float* __restrict__ alpha, const float* __restrict__ tval,
    const float* __restrict__ rgb, const float* __restrict__ nab,
    const int* __restrict__ rays, const int* __restrict__ segStart,
    const int* __restrict__ segEnd, float* __restrict__ out,
    int numHit, int numRays, int T) {
  __shared__ float sA[CAP_SAMPLES];
  __shared__ float sT[CAP_SAMPLES];
  __shared__ int sLo, sHi;

  const int s = blockIdx.x * blockDim.x + threadIdx.x;
  int st = 0, en = 0;
  if (s < numHit) { st = segStart[s]; en = segEnd[s]; }

  if (threadIdx.x == 0) { sLo = 0x7fffffff; sHi = 0; }
  __syncthreads();
  if (en > st) { atomicMin(&sLo, st); atomicMax(&sHi, en); }
  __syncthreads();
  const int lo = sLo, hi = sHi;
  const int base = lo & ~3;                              // 16B-aligned staging base
  const bool useLds = (hi > lo) && ((hi - base) <= CAP_SAMPLES);

  if (useLds) {
#if HAVE_ASYNC_B128
    const int n16 = (hi - base + 3) >> 2;                // 16B chunks to stage
    for (int c = (int)threadIdx.x; c < n16; c += (int)blockDim.x) {
      const int idx = base + 4 * c;
      if (idx + 4 <= T) {
        __builtin_amdgcn_global_load_async_to_lds_b128(
            (__AS1 v4i_t*)(alpha + idx), (__AS3 v4i_t*)(sA + 4 * c), 0, 0);
        __builtin_amdgcn_global_load_async_to_lds_b128(
            (__AS1 v4i_t*)(tval + idx), (__AS3 v4i_t*)(sT + 4 * c), 0, 0);
      } else {                                           // ragged tail at end of buffer
        for (int j = idx; j < hi; ++j) {
#if HAVE_ASYNC_B32
          __builtin_amdgcn_global_load_async_to_lds_b32(
              (__AS1 int*)(alpha + j), (__AS3 int*)(sA + (j - base)), 0, 0);
          __builtin_amdgcn_global_load_async_to_lds_b32(
              (__AS1 int*)(tval + j), (__AS3 int*)(sT + (j - base)), 0, 0);
#else
          sA[j - base] = alpha[j];
          sT[j - base] = tval[j];
#endif
        }
      }
    }
    svr_wait_async0();
#else   // no async builtins: plain cooperative staging
    for (int i = lo + (int)threadIdx.x; i < hi; i += (int)blockDim.x) {
      sA[i - base] = alpha[i];
      sT[i - base] = tval[i];
    }
#endif
    __syncthreads();
  }

  float Tr = 1.0f, vwsum = 0.0f, svt = 0.0f;
  float r0 = 0.0f, r1 = 0.0f, r2 = 0.0f;
  float n0 = 0.0f, n1 = 0.0f, n2 = 0.0f;

  for (int i = st; i < en; ++i) {
    const float a  = useLds ? sA[i - base] : alpha[i];
    const float ti = useLds ? sT[i - base] : tval[i];
    const float vw = a * Tr;
    Tr *= (1.0f - a);
    vwsum += vw;
    svt += vw * ti;

    const float cr = __builtin_nontemporal_load(rgb + 3 * i + 0);
    const float cg = __builtin_nontemporal_load(rgb + 3 * i + 1);
    const float cb = __builtin_nontemporal_load(rgb + 3 * i + 2);
    r0 += vw * cr; r1 += vw * cg; r2 += vw * cb;

    float nx = __builtin_nontemporal_load(nab + 3 * i + 0);
    float ny = __builtin_nontemporal_load(nab + 3 * i + 1);
    float nz = __builtin_nontemporal_load(nab + 3 * i + 2);
    nx = fminf(fmaxf(nx, -1.0f), 1.0f);
    ny = fminf(fmaxf(ny, -1.0f), 1.0f);
    nz = fminf(fmaxf(nz, -1.0f), 1.0f);
    const float L = sqrtf(nx * nx + ny * ny + nz * nz);
    const float inv = 1.0f / fmaxf(L, 1e-12f);
    n0 += vw * nx * inv; n1 += vw * ny * inv; n2 += vw * nz * inv;
  }

  if (s < numHit) {
    const int r = rays[s];
    __builtin_nontemporal_store(vwsum, out + r);                              // mask_volume
    __builtin_nontemporal_store(svt / (vwsum + 1e-10f), out + numRays + r);   // depth_volume
    float* __restrict__ rgbo = out + 2 * numRays;                             // rgb_volume
    __builtin_nontemporal_store(r0, rgbo + 3 * r + 0);
    __builtin_nontemporal_store(r1, rgbo + 3 * r + 1);
    __builtin_nontemporal_store(r2, rgbo + 3 * r + 2);
    float* __restrict__ nrmo = out + 5 * numRays;                             // normals_volume
    __builtin_nontemporal_store(n0, nrmo + 3 * r + 0);
    __builtin_nontemporal_store(n1, nrmo + 3 * r + 1);
    __builtin_nontemporal_store(n2, nrmo + 3 * r + 2);
  }
}

extern "C" void kernel_launch(void* const* d_in, const int* in_sizes, int n_in,
                              void* d_out, int out_size, void* d_ws, size_t ws_size,
                              hipStream_t stream) {
  const float* alpha = (const float*)d_in[0];
  const float* tval  = (const float*)d_in[1];
  const float* rgb   = (const float*)d_in[2];
  const float* nab   = (const float*)d_in[3];
  const int*   rays  = (const int*)d_in[4];
  const int*   seg   = (const int*)d_in[5];

  const int T       = in_sizes[0];      // total packed samples
  const int numHit  = in_sizes[4];      // number of hit rays (segments)
  const int numRays = out_size / 8;     // out = [N] + [N] + [N,3] + [N,3]

  int* segStart = (int*)d_ws;           // [numHit]
  int* segEnd   = segStart + numHit;    // [numHit]
  (void)ws_size; (void)n_in;

  float* out = (float*)d_out;

  svr_zero<<<2048, 256, 0, stream>>>(out, out_size, segStart, 2 * numHit);
  svr_seg_bounds<<<(T + 255) / 256, 256, 0, stream>>>(seg, T, segStart, segEnd);
  svr_render<<<(numHit + 255) / 256, 256, 0, stream>>>(
      alpha, tval, rgb, nab, rays, segStart, segEnd, out, numHit, numRays, T);
}